// Legendre_12704513261900
// MI455X (gfx1250) — compile-verified
//
#include <hip/hip_runtime.h>

#define BASIS 64
#define TPB   256
#define ROW_V 17   // LDS row stride in float4s: 16 data + 1 pad -> 272 B (16B aligned, bank-skewed)

#if defined(__has_builtin)
#  if __has_builtin(__builtin_amdgcn_global_store_async_from_lds_b128)
#    define HAVE_ASYNC128 1
#  endif
#  if __has_builtin(__builtin_amdgcn_s_wait_dscnt)
#    define HAVE_WAIT_DS 1
#  endif
#  if __has_builtin(__builtin_amdgcn_s_wait_asynccnt)
#    define HAVE_WAIT_ASYNC 1
#  endif
#endif

// Async builtin params are int4* in AS1 (global) / AS3 (LDS).
typedef int v4i __attribute__((ext_vector_type(4)));
typedef __attribute__((address_space(1))) v4i* gv4_ptr;
typedef __attribute__((address_space(3))) v4i* lv4_ptr;

#define STORE_CPOL 1  // TH_NT: non-temporal streaming store (output >> L2)

__device__ __forceinline__ void wait_dscnt0() {
#if defined(HAVE_WAIT_DS)
    __builtin_amdgcn_s_wait_dscnt(0);
#else
    asm volatile("s_wait_dscnt 0x0" ::: "memory");
#endif
}

__device__ __forceinline__ void wait_asynccnt0() {
#if defined(HAVE_WAIT_ASYNC)
    __builtin_amdgcn_s_wait_asynccnt(0);
#else
    asm volatile("s_wait_asynccnt 0x0" ::: "memory");
#endif
}

__device__ __forceinline__ void store_chunk(float4* lds4, float* blockOut, int cid) {
    const int row = cid >> 4;   // t-row within block
    const int cr  = cid & 15;   // 16B chunk within row
    float* gdst  = blockOut + (size_t)cid * 4;
    float4* lsrc = &lds4[row * ROW_V + cr];
#if defined(HAVE_ASYNC128)
    __builtin_amdgcn_global_store_async_from_lds_b128(
        (gv4_ptr)gdst, (lv4_ptr)lsrc, /*offset=*/0, /*cpol=*/STORE_CPOL);
#else
    *(float4*)gdst = *lsrc;
#endif
}

__global__ __launch_bounds__(TPB) void legendre_stream_kernel(
    const float* __restrict__ t, const float* __restrict__ arange,
    float* __restrict__ out, int n)
{
    __shared__ float4 lds4[TPB * ROW_V];  // 69,632 B, naturally 16B-aligned

    const int tid       = threadIdx.x;
    const int blockBase = blockIdx.x * TPB;
    const int gid       = blockBase + tid;

    // sqrt(2k+1) scales: wave-uniform address -> SGPRs via s_load
    float s[BASIS];
#pragma unroll
    for (int k = 0; k < BASIS; ++k) s[k] = arange[k];

    // x = 2t - 1
    const float x = (gid < n) ? (2.0f * t[gid] - 1.0f) : 0.0f;

    // Three-term Legendre recurrence, scaled, staged into LDS as b128 chunks.
    //   P_k = ((2k-1)/k) x P_{k-1} - ((k-1)/k) P_{k-2}
    float pm1 = 1.0f;   // P_0
    float p   = x;      // P_1
#pragma unroll
    for (int c = 0; c < BASIS / 4; ++c) {
        float4 v;
#pragma unroll
        for (int j = 0; j < 4; ++j) {
            const int k = 4 * c + j;
            float val;
            if (k == 0) {
                val = 1.0f;
            } else if (k == 1) {
                val = x;
            } else {
                const float A = (float)(2 * k - 1) / (float)k;  // constant-folded
                const float B = (float)(k - 1) / (float)k;      // constant-folded
                const float pn = A * x * p - B * pm1;
                pm1 = p;
                p   = pn;
                val = pn;
            }
            (&v.x)[j] = s[k] * val;
        }
        lds4[tid * ROW_V + c] = v;  // single ds_store_b128
    }

    // Wave-synchronous: this wave's 16 ds_store_b128 cover all 32 of its rows;
    // drain DS before the DMA engine reads LDS.
    wait_dscnt0();

    // Each wave streams out its own 32 rows (8 KB) as 16B chunks,
    // lane-contiguous in global memory (512 B per issued instruction).
    float* blockOut = out + (size_t)blockBase * BASIS;
    const int w    = tid >> 5;
    const int lane = tid & 31;

    if (blockBase + TPB <= n) {
        // Full block: uniform, no per-chunk guard (hot path; n = 2^21 here).
#pragma unroll
        for (int c = 0; c < BASIS / 4; ++c)
            store_chunk(lds4, blockOut, (w << 9) + (c << 5) + lane);
    } else {
        // Tail block: guard per chunk.
#pragma unroll
        for (int c = 0; c < BASIS / 4; ++c) {
            const int cid = (w << 9) + (c << 5) + lane;
            if (blockBase + (cid >> 4) < n)
                store_chunk(lds4, blockOut, cid);
        }
    }

    // LDS must stay valid until the DMA drains.
    wait_asynccnt0();
}

extern "C" void kernel_launch(void* const* d_in, const int* in_sizes, int n_in,
                              void* d_out, int out_size, void* d_ws, size_t ws_size,
                              hipStream_t stream) {
    const float* t      = (const float*)d_in[0];
    const float* arange = (const float*)d_in[1];
    float*       out    = (float*)d_out;
    const int n = in_sizes[0];

    const int blocks = (n + TPB - 1) / TPB;
    legendre_stream_kernel<<<blocks, TPB, 0, stream>>>(t, arange, out, n);
}